// PredictionHead_65137474011193
// MI455X (gfx1250) — compile-verified
//
#include <hip/hip_runtime.h>
#include <hip/hip_bf16.h>
#include <stdint.h>

#define T_     8
#define N_     20000
#define INDIM_ 256
#define EMB_   128
#define E_     200000

typedef __attribute__((ext_vector_type(2))) float v2f;
typedef __attribute__((ext_vector_type(4))) float v4f;
typedef __attribute__((ext_vector_type(8))) float v8f;

// -------- Phase 1: h = relu(x @ Wp^T + bp) via V_WMMA_F32_16X16X4_F32 --------
// x: [T*N, 256] row-major, Wp: [128, 256] row-major, h: [T*N, 128]
// 4 waves / WG; Wp staged in LDS (128 KB of the 320 KB WGP pool).
// Each wave: one 16-row M block, all 8 N-tiles accumulated in 64 VGPRs.
__global__ __launch_bounds__(128) void proj_relu_wmma(
    const float* __restrict__ x, const float* __restrict__ Wp,
    const float* __restrict__ bp, float* __restrict__ h)
{
    __shared__ float wpl[EMB_ * INDIM_];   // 128*256*4 B = 128 KB

    // cooperative Wp -> LDS (b128 transfers)
    for (int i = threadIdx.x; i < (EMB_ * INDIM_) / 4; i += 128)
        ((v4f*)wpl)[i] = ((const v4f*)Wp)[i];
    __syncthreads();

    const int lane  = threadIdx.x & 31;
    const int wave  = threadIdx.x >> 5;
    const int row0  = (blockIdx.x * 4 + wave) * 16;
    const int mrow  = lane & 15;    // M row within tile (A), N col within tile (B/C/D)
    const int khalf = lane >> 4;    // 0: K{0,1}, 1: K{2,3}  (32-bit A-frag layout)

    v8f acc[8];
#pragma unroll
    for (int n = 0; n < 8; ++n) acc[n] = 0.0f;

    const float* xrow = x   + (size_t)(row0 + mrow) * INDIM_ + khalf * 2;
    const float* wrow = wpl + (size_t)mrow * INDIM_ + khalf * 2;

    for (int k = 0; k < INDIM_; k += 4) {
        v2f a = *(const v2f*)(xrow + k);               // A frag: 16x4 f32
#pragma unroll
        for (int n = 0; n < 8; ++n) {
            v2f b = *(const v2f*)(wrow + n * 16 * INDIM_ + k);  // B frag: 4x16 (ds_load_b64)
            acc[n] = __builtin_amdgcn_wmma_f32_16x16x4_f32(
                false, a, false, b, (short)0, acc[n], false, false);
        }
    }

    // epilogue: bias + relu + store (C layout: vgpr r, lane -> M=r+8*khalf, N=mrow)
#pragma unroll
    for (int n = 0; n < 8; ++n) {
        const int   ncol = n * 16 + mrow;
        const float bias = bp[ncol];
#pragma unroll
        for (int r = 0; r < 8; ++r) {
            float v = acc[n][r] + bias;
            v = v > 0.0f ? v : 0.0f;
            h[(size_t)(row0 + r + khalf * 8) * EMB_ + ncol] = v;
        }
    }
}

// -------- Phase 2: per-pair masked |diff|.Wc score, mean over frames --------
// One wave per pair; lane covers 4 of 128 dims (b128 loads, L2-resident h).
// num = sum_t m_t*dot_t accumulated lane-locally; single shfl reduction/pair.
__global__ __launch_bounds__(256) void pair_score(
    const float* __restrict__ h, const float* __restrict__ psmask,
    const long long* __restrict__ eoi, const float* __restrict__ Wc,
    const float* __restrict__ bc, float* __restrict__ out)
{
    const int lane = threadIdx.x & 31;
    const int wave = threadIdx.x >> 5;
    const int e    = blockIdx.x * 8 + wave;
    if (e >= E_) return;

    const int ia = (int)eoi[e];
    const int ib = (int)eoi[E_ + e];
    const int kb = lane * 4;

    const v4f wc = *(const v4f*)(Wc + kb);

    float num = 0.0f, den = 0.0f;
#pragma unroll
    for (int t = 0; t < T_; ++t) {
        const v4f ha = *(const v4f*)(h + ((size_t)t * N_ + ia) * EMB_ + kb);
        const v4f hb = *(const v4f*)(h + ((size_t)t * N_ + ib) * EMB_ + kb);
        float p = fabsf(ha.x - hb.x) * wc.x;
        p = fmaf(fabsf(ha.y - hb.y), wc.y, p);
        p = fmaf(fabsf(ha.z - hb.z), wc.z, p);
        p = fmaf(fabsf(ha.w - hb.w), wc.w, p);
        const float m = psmask[t * N_ + ia] * psmask[t * N_ + ib];
        num = fmaf(m, p, num);
        den += m;
    }

#pragma unroll
    for (int off = 16; off > 0; off >>= 1)
        num += __shfl_xor(num, off, 32);

    if (lane == 0)
        out[e] = (num + bc[0] * den) / (den + 1e-4f);
}

extern "C" void kernel_launch(void* const* d_in, const int* in_sizes, int n_in,
                              void* d_out, int out_size, void* d_ws, size_t ws_size,
                              hipStream_t stream) {
    const float*     x      = (const float*)d_in[0];
    const float*     psmask = (const float*)d_in[1];
    const long long* eoi    = (const long long*)d_in[2];   // int64 in reference
    const float*     Wp     = (const float*)d_in[3];
    const float*     bp     = (const float*)d_in[4];
    const float*     Wc     = (const float*)d_in[5];
    const float*     bc     = (const float*)d_in[6];
    float* out = (float*)d_out;
    float* h   = (float*)d_ws;   // 8*20000*128 floats = 81.92 MB (fits L2: 192 MB)

    // Phase 1: 160000 rows / (16 rows/wave * 4 waves/WG) = 2500 WGs
    proj_relu_wmma<<<dim3(2500), dim3(128), 0, stream>>>(x, Wp, bp, h);
    // Phase 2: 200000 pairs / 8 waves per WG = 25000 WGs
    pair_score<<<dim3(25000), dim3(256), 0, stream>>>(h, psmask, eoi, Wc, bc, out);
}